// SE3BNConvolution_25237227831400
// MI455X (gfx1250) — compile-verified
//
#include <hip/hip_runtime.h>

// ---------------------------------------------------------------------------
// SE3 BN + dynamic-kernel 3D conv for MI455X (gfx1250, wave32, WMMA).
// Implicit GEMM through V_WMMA_F32_16X16X4_F32 (fp32 matrix pipe), with
// double-buffered GLOBAL_LOAD_ASYNC_TO_LDS staging of the dynamic kernel.
// ---------------------------------------------------------------------------

typedef __attribute__((ext_vector_type(2))) float v2f;
typedef __attribute__((ext_vector_type(8))) float v8f;

#define EPS_BN     1e-5f
#define NB         8                 // batch slices for stats partials
#define CIN        72
#define SPATIAL    32768             // 32^3
#define NCH_STRIDE (CIN * SPATIAL)   // per-batch stride in floats
#define WOUT       30
#define OSPATIAL   27000             // 30^3
#define NTAP       27
#define KB_COUNT   18                // 72 / 4
#define NPAD       80                // 72 padded to 5 N-tiles of 16
#define TAPSLICE   (KB_COUNT * 2 * NPAD * 2)   // 5760 floats per tap

// workspace layout (float offsets)
#define WS_PART  0        // 72 * NB * 2
#define WS_MEAN  1152     // 8
#define WS_SCALE 1160     // 72 (per-input-channel BN scale)
#define WS_BIAS  1232     // 80 (zero beyond channel 7)
#define WS_KSWZ  1312     // 27 * 5760 = 155520  (16B aligned: 1312*4 = 5248)

// ---------------------------------------------------------------------------
// Async copy: 16B per lane, global -> LDS, tracked by ASYNCcnt.
// vdst = per-lane LDS byte address (flat-addr[31:0] of a __shared__ pointer,
// per CDNA5 ISA 10.2 the low 32 bits are the LDS logical address).
// ---------------------------------------------------------------------------
__device__ __forceinline__ void async_load_f4(unsigned lds_off, const float* g) {
    asm volatile("global_load_async_to_lds_b128 %0, %1, off"
                 :: "v"(lds_off), "v"((unsigned long long)(uintptr_t)g)
                 : "memory");
}
__device__ __forceinline__ void wait_async0() {
    asm volatile("s_wait_asynccnt 0x0" ::: "memory");
}

// ---------------------------------------------------------------------------
// Kernel 1: per-(channel, batch-slice) sum and sum-of-squares partials.
// ---------------------------------------------------------------------------
__global__ void se3_stats_partial(const float* __restrict__ x,
                                  float* __restrict__ ws) {
    const int c = blockIdx.x >> 3;   // channel 0..71
    const int n = blockIdx.x & 7;    // batch    0..7
    const float* p = x + (size_t)n * NCH_STRIDE + (size_t)c * SPATIAL;
    float s = 0.f, q = 0.f;
    for (int i = threadIdx.x; i < SPATIAL; i += 256) {
        float v = p[i];
        s += v;
        q += v * v;
    }
    __shared__ float sh_s[256], sh_q[256];
    sh_s[threadIdx.x] = s;
    sh_q[threadIdx.x] = q;
    __syncthreads();
    for (int off = 128; off > 0; off >>= 1) {
        if ((int)threadIdx.x < off) {
            sh_s[threadIdx.x] += sh_s[threadIdx.x + off];
            sh_q[threadIdx.x] += sh_q[threadIdx.x + off];
        }
        __syncthreads();
    }
    if (threadIdx.x == 0) {
        ws[WS_PART + (c * NB + n) * 2 + 0] = sh_s[0];
        ws[WS_PART + (c * NB + n) * 2 + 1] = sh_q[0];
    }
}

// ---------------------------------------------------------------------------
// Kernel 2: finalize stats -> per-channel scales, scalar means, bias.
// Fields: ch 0..7 (m=8,d=1), ch 8..31 (m=8,d=3), ch 32..71 (m=8,d=5).
// ---------------------------------------------------------------------------
__global__ void se3_stats_final(const float* __restrict__ weight,
                                const float* __restrict__ basis00,
                                float* __restrict__ ws) {
    __shared__ float sh_mean[72], sh_e2[72], sh_sc0[8], sh_m0[8];
    const int t = threadIdx.x;      // 128 threads
    if (t < 72) {
        float s = 0.f, q = 0.f;
        for (int n = 0; n < NB; ++n) {
            s += ws[WS_PART + (t * NB + n) * 2 + 0];
            q += ws[WS_PART + (t * NB + n) * 2 + 1];
        }
        const float inv = 1.0f / (float)(NB * SPATIAL);
        sh_mean[t] = s * inv;
        sh_e2[t]   = q * inv;
    }
    __syncthreads();
    if (t < 8) {
        // scalar field: centered variance
        float m0  = sh_mean[t];
        float var = sh_e2[t] - m0 * m0;
        float sc0 = rsqrtf(var + EPS_BN);
        sh_m0[t]  = m0;
        sh_sc0[t] = sc0;
        ws[WS_MEAN + t]  = m0;
        ws[WS_SCALE + t] = sc0;
        // d=3 field: norm = sum_d E[x^2]
        float n1 = sh_e2[8 + 3 * t] + sh_e2[9 + 3 * t] + sh_e2[10 + 3 * t];
        float sc1 = rsqrtf(n1 + EPS_BN);
        ws[WS_SCALE + 8 + 3 * t + 0] = sc1;
        ws[WS_SCALE + 8 + 3 * t + 1] = sc1;
        ws[WS_SCALE + 8 + 3 * t + 2] = sc1;
        // d=5 field
        float n2 = 0.f;
        for (int d = 0; d < 5; ++d) n2 += sh_e2[32 + 5 * t + d];
        float sc2 = rsqrtf(n2 + EPS_BN);
        for (int d = 0; d < 5; ++d) ws[WS_SCALE + 32 + 5 * t + d] = sc2;
    }
    __syncthreads();
    if (t < 80) {
        float b = 0.f;
        if (t < 8) {
            // identity[b] = sum over all 27 taps of basis_0_0[b,0,0,:]
            float id0 = 0.f, id1 = 0.f;
            for (int k = 0; k < 27; ++k) { id0 += basis00[k]; id1 += basis00[27 + k]; }
            for (int v = 0; v < 8; ++v) {
                float corr = (weight[(t * 8 + v) * 2 + 0] * id0 +
                              weight[(t * 8 + v) * 2 + 1] * id1) * sh_sc0[v];
                b -= corr * sh_m0[v];
            }
        }
        ws[WS_BIAS + t] = b;
    }
}

// ---------------------------------------------------------------------------
// Kernel 3: build dynamic conv kernel, pre-swizzled for the WMMA B layout.
//   Kswz[tap][kb][half][co(80)][pair]  with ci = 4*kb + 2*half + pair.
// ---------------------------------------------------------------------------
__global__ void se3_build_kernel(const float* __restrict__ weight,
                                 const float* __restrict__ scale_ch,
                                 const float* __restrict__ b00, const float* __restrict__ b01,
                                 const float* __restrict__ b02, const float* __restrict__ b10,
                                 const float* __restrict__ b11, const float* __restrict__ b12,
                                 const float* __restrict__ b20, const float* __restrict__ b21,
                                 const float* __restrict__ b22,
                                 float* __restrict__ kswz) {
    int f = blockIdx.x * 256 + threadIdx.x;
    if (f >= NTAP * TAPSLICE) return;
    int idx = f;
    const int p  = idx & 1;  idx >>= 1;
    const int co = idx % NPAD; idx /= NPAD;
    const int h  = idx & 1;  idx >>= 1;
    const int kb = idx % KB_COUNT; idx /= KB_COUNT;
    const int t  = idx;                       // 0..26
    float val = 0.f;
    if (co < 72) {
        const int ci = kb * 4 + h * 2 + p;
        int i, u, dm, di, j, v, e, dj;
        if (co < 8)       { i = 0; u = co;            dm = 0;             di = 1; }
        else if (co < 32) { i = 1; u = (co - 8) / 3;  dm = (co - 8) % 3;  di = 3; }
        else              { i = 2; u = (co - 32) / 5; dm = (co - 32) % 5; di = 5; }
        if (ci < 8)       { j = 0; v = ci;            e  = 0;             dj = 1; }
        else if (ci < 32) { j = 1; v = (ci - 8) / 3;  e  = (ci - 8) % 3;  dj = 3; }
        else              { j = 2; v = (ci - 32) / 5; e  = (ci - 32) % 5; dj = 5; }
        const float* bs[9] = {b00, b01, b02, b10, b11, b12, b20, b21, b22};
        const float* basis = bs[i * 3 + j];
        const float* w = weight + (i * 3 + j) * 128 + (u * 8 + v) * 2;
        float acc = 0.f;
        for (int b = 0; b < 2; ++b)
            acc += w[b] * basis[((b * di + dm) * dj + e) * 27 + t];
        val = acc * scale_ch[ci];
    }
    kswz[f] = val;
}

// ---------------------------------------------------------------------------
// Kernel 4: implicit-GEMM conv via V_WMMA_F32_16X16X4_F32.
//   Per wave: 32 outputs (two adjacent y-rows x 16 x-positions) x 80 channels,
//   K = 72 ch * 27 taps. Per K-block: 4 vmem + 5 ds feed 10 WMMAs.
//   A layout: lanes 0-15 (M), VGPR0/1 = K0/K1; lanes 16-31 = K2/K3.
//   B layout: lanes 0-15 = N, VGPR0/1 = K0/K1; lanes 16-31 = K2/K3.
//   C layout: lane = (N, M-half), VGPR r = M row -> 8 consecutive x stores.
//   Dynamic kernel double-buffered in LDS via async global->LDS DMA.
// ---------------------------------------------------------------------------
__global__ void __launch_bounds__(128)
se3_conv_wmma(const float* __restrict__ x,
              const float* __restrict__ kswz,
              const float* __restrict__ bias,
              float* __restrict__ out) {
    __shared__ __align__(16) float bsh[2 * TAPSLICE];   // 2 x 23040 B

    const int tid  = threadIdx.x;
    const int lane = tid & 31;
    const int wv   = tid >> 5;             // wave in block: 0..3
    const int l15  = lane & 15;
    const int h    = lane >> 4;

    int wid = blockIdx.x * 4 + wv;         // 0..7199
    const int tx = wid & 1;                // x tile: 0 -> x0=0, 1 -> x0=14
    int tmp = wid >> 1;
    const int yp = tmp % 15; tmp /= 15;    // y-row pair
    const int z  = tmp % WOUT; tmp /= WOUT;
    const int n  = tmp;                    // 0..7
    const int x0 = tx ? 14 : 0;
    const int y0 = yp * 2;

    v8f acc0[5] = {};                      // row y0
    v8f acc1[5] = {};                      // row y0+1

    const float* xn = x + (size_t)n * NCH_STRIDE;
    const int aoff_lane = l15 + h * (2 * SPATIAL);   // M index + channel pair
    const int boff_lane = (h * NPAD + l15) * 2;

    const unsigned lds0 = (unsigned)(uintptr_t)(&bsh[0]);

    // prologue: async-stage tap 0 into buffer 0 (1440 float4s / 128 threads)
    for (int i = tid; i < TAPSLICE / 4; i += 128)
        async_load_f4(lds0 + (unsigned)i * 16u, kswz + i * 4);

    for (int tap = 0; tap < NTAP; ++tap) {
        wait_async0();        // my quarter of buf[tap&1] has landed
        __syncthreads();      // everyone's landed; previous readers done

        if (tap + 1 < NTAP) { // overlap next tap's DMA with this tap's WMMAs
            const float* g = kswz + (size_t)(tap + 1) * TAPSLICE;
            const unsigned dst = lds0 + (unsigned)(((tap + 1) & 1) * TAPSLICE * 4);
            for (int i = tid; i < TAPSLICE / 4; i += 128)
                async_load_f4(dst + (unsigned)i * 16u, g + i * 4);
        }

        const int dz = tap / 9, rem = tap % 9, dy = rem / 3, dxo = rem % 3;
        const float* xr0 = xn + (size_t)((z + dz) * 32 + (y0 + dy)) * 32
                              + x0 + dxo + aoff_lane;
        const float* xr1 = xr0 + 32;                       // y0+1 input row
        const float* bp  = bsh + (tap & 1) * TAPSLICE + boff_lane;

#pragma unroll 2
        for (int kb = 0; kb < KB_COUNT; ++kb) {
            v2f a0, a1;
            a0.x = xr0[0];        // ci = 4*kb + 2*h
            a0.y = xr0[SPATIAL];  // ci + 1
            a1.x = xr1[0];
            a1.y = xr1[SPATIAL];
#pragma unroll
            for (int nt = 0; nt < 5; ++nt) {
                v2f b = *(const v2f*)(bp + nt * 32);
                acc0[nt] = __builtin_amdgcn_wmma_f32_16x16x4_f32(
                    false, a0, false, b, (short)0, acc0[nt], false, false);
                acc1[nt] = __builtin_amdgcn_wmma_f32_16x16x4_f32(
                    false, a1, false, b, (short)0, acc1[nt], false, false);
            }
            xr0 += 4 * SPATIAL;
            xr1 += 4 * SPATIAL;
            bp  += 2 * NPAD * 2;
        }
    }

    // store: lane (l15,h) VGPR r -> out[n, co=nt*16+l15, z, y, x0+8h+r] + bias
#pragma unroll
    for (int nt = 0; nt < 5; ++nt) {
        const int co = nt * 16 + l15;
        if (co < 72) {
            const float bia = bias[co];
            float* op = out + ((size_t)n * 72 + co) * OSPATIAL
                            + (size_t)z * 900 + (size_t)y0 * 30 + x0 + 8 * h;
#pragma unroll
            for (int r = 0; r < 8; ++r) {
                op[r]      = acc0[nt][r] + bia;   // row y0
                op[30 + r] = acc1[nt][r] + bia;   // row y0+1
            }
        }
    }
}

// ---------------------------------------------------------------------------
extern "C" void kernel_launch(void* const* d_in, const int* in_sizes, int n_in,
                              void* d_out, int out_size, void* d_ws, size_t ws_size,
                              hipStream_t stream) {
    (void)in_sizes; (void)n_in; (void)out_size; (void)ws_size;
    const float* x      = (const float*)d_in[0];
    const float* weight = (const float*)d_in[1];
    const float* b00 = (const float*)d_in[2];
    const float* b01 = (const float*)d_in[3];
    const float* b02 = (const float*)d_in[4];
    const float* b10 = (const float*)d_in[5];
    const float* b11 = (const float*)d_in[6];
    const float* b12 = (const float*)d_in[7];
    const float* b20 = (const float*)d_in[8];
    const float* b21 = (const float*)d_in[9];
    const float* b22 = (const float*)d_in[10];
    float* ws  = (float*)d_ws;
    float* out = (float*)d_out;

    se3_stats_partial<<<dim3(72 * NB), dim3(256), 0, stream>>>(x, ws);
    se3_stats_final<<<dim3(1), dim3(128), 0, stream>>>(weight, b00, ws);
    se3_build_kernel<<<dim3((NTAP * TAPSLICE + 255) / 256), dim3(256), 0, stream>>>(
        weight, ws + WS_SCALE, b00, b01, b02, b10, b11, b12, b20, b21, b22,
        ws + WS_KSWZ);
    // 7200 M-tiles (8 batch x 30 z x 15 y-pairs x 2 x-tiles), 4 waves / block
    se3_conv_wmma<<<dim3(1800), dim3(128), 0, stream>>>(
        x, ws + WS_KSWZ, ws + WS_BIAS, out);
}